// Gate_38199439130700
// MI455X (gfx1250) — compile-verified
//
#include <hip/hip_runtime.h>
#include <hip/hip_bf16.h>
#include <math.h>

typedef float v2f __attribute__((ext_vector_type(2)));
typedef float v8f __attribute__((ext_vector_type(8)));

#define BS_    16384
#define XD_    4096
#define KD_    512
#define NK_    4096
#define TOPK_  16

// ---------------- GEMM: C[M,N] = A[M,K] * B[N,K]^T (+ bias[n]) ----------------
// Block tile 128x128, K-chunk 16, 256 threads = 8 waves (wave32).
// Waves arranged 4(M) x 2(N); each wave computes a 32x64 patch = 2x4 WMMA tiles.
// Global -> LDS staging uses gfx1250 async-to-LDS (ASYNCcnt) with double buffering.
constexpr int BM  = 128;
constexpr int BN  = 128;
constexpr int KT  = 16;
constexpr int PAD = 20;   // LDS pitch (floats): 16B-aligned rows, conflict-free frag reads

// Low 32 bits of a flat shared-memory address == LDS byte address (ISA flat aperture rule).
__device__ __forceinline__ uint32_t lds_u32(const void* p) {
  return (uint32_t)(uintptr_t)p;
}

__device__ __forceinline__ void async_copy_b128(uint32_t lds_byte_addr, const float* gptr) {
  asm volatile("global_load_async_to_lds_b128 %0, %1, off"
               :: "v"(lds_byte_addr), "v"(gptr)
               : "memory");
}

__global__ __launch_bounds__(256)
void gemm_tn_wmma(const float* __restrict__ A, const float* __restrict__ B,
                  const float* __restrict__ bias, float* __restrict__ C,
                  int M, int N, int K) {
  __shared__ float lA[2][BM * PAD];
  __shared__ float lB[2][BN * PAD];

  const int tid  = threadIdx.x;
  const int wave = tid >> 5;
  const int lane = tid & 31;
  const int half = lane >> 4;   // 0: lanes 0-15, 1: lanes 16-31
  const int l16  = lane & 15;

  const int bm = blockIdx.x * BM;
  const int bn = blockIdx.y * BN;

  const int wm = (wave & 3) * 32;   // wave row offset in block tile (0,32,64,96)
  const int wn = (wave >> 2) * 64;  // wave col offset in block tile (0,64)

  v8f acc[2][4];
  const v8f vzero = {};
#pragma unroll
  for (int i = 0; i < 2; ++i)
#pragma unroll
    for (int j = 0; j < 4; ++j)
      acc[i][j] = vzero;

  // Cooperative async-load mapping: each thread owns one 16B chunk of 4 rows.
  const int arow = tid >> 2;          // 0..63
  const int acg  = (tid & 3) * 4;     // float offset within KT chunk

  const float* gA0 = A + (size_t)(bm + arow)      * K + acg;
  const float* gA1 = A + (size_t)(bm + arow + 64) * K + acg;
  const float* gB0 = B + (size_t)(bn + arow)      * K + acg;
  const float* gB1 = B + (size_t)(bn + arow + 64) * K + acg;

  uint32_t dA0[2], dA1[2], dB0[2], dB1[2];
#pragma unroll
  for (int s = 0; s < 2; ++s) {
    dA0[s] = lds_u32(&lA[s][arow        * PAD + acg]);
    dA1[s] = lds_u32(&lA[s][(arow + 64) * PAD + acg]);
    dB0[s] = lds_u32(&lB[s][arow        * PAD + acg]);
    dB1[s] = lds_u32(&lB[s][(arow + 64) * PAD + acg]);
  }

  // Prologue: async-fetch chunk 0 into buffer 0.
  async_copy_b128(dA0[0], gA0);
  async_copy_b128(dA1[0], gA1);
  async_copy_b128(dB0[0], gB0);
  async_copy_b128(dB1[0], gB1);

  int buf = 0;
  for (int k0 = 0; k0 < K; k0 += KT) {
    if (k0 + KT < K) {
      // Fetch next chunk into the other buffer (safe: last barrier cleared its readers),
      // then wait for the *current* chunk's 4 async ops (in-order per wave).
      const int nb = buf ^ 1;
      async_copy_b128(dA0[nb], gA0 + k0 + KT);
      async_copy_b128(dA1[nb], gA1 + k0 + KT);
      async_copy_b128(dB0[nb], gB0 + k0 + KT);
      async_copy_b128(dB1[nb], gB1 + k0 + KT);
      asm volatile("s_wait_asynccnt 0x4" ::: "memory");
    } else {
      asm volatile("s_wait_asynccnt 0x0" ::: "memory");
    }
    __syncthreads();   // all waves' async data for this chunk visible in LDS

    const float* cA = lA[buf];
    const float* cB = lB[buf];
#pragma unroll
    for (int kk = 0; kk < KT; kk += 4) {
      // A frag 16x4 f32: v0={K=kk | K=kk+2}, v1={K=kk+1 | K=kk+3} split at lane 16.
      v2f af[2], bf[4];
#pragma unroll
      for (int i = 0; i < 2; ++i)
        af[i] = *(const v2f*)&cA[(wm + 16 * i + l16) * PAD + kk + 2 * half];
#pragma unroll
      for (int j = 0; j < 4; ++j)
        bf[j] = *(const v2f*)&cB[(wn + 16 * j + l16) * PAD + kk + 2 * half];

#pragma unroll
      for (int i = 0; i < 2; ++i)
#pragma unroll
        for (int j = 0; j < 4; ++j)
          acc[i][j] = __builtin_amdgcn_wmma_f32_16x16x4_f32(
              false, af[i], false, bf[j], (short)0, acc[i][j], false, false);
    }
    __syncthreads();   // readers done before next overwrite of buf^1... (next iter writes buf)
    buf ^= 1;
  }

  // C/D layout: VGPR e -> rows (e, e+8) split at lane 16; lanes 0-15 are N=l16.
#pragma unroll
  for (int i = 0; i < 2; ++i) {
#pragma unroll
    for (int j = 0; j < 4; ++j) {
      const int col = bn + wn + 16 * j + l16;
      const float bb = bias ? bias[col] : 0.0f;
#pragma unroll
      for (int e = 0; e < 8; ++e) {
        const int row = bm + wm + 16 * i + e + 8 * half;
        C[(size_t)row * N + col] = acc[i][j][e] + bb;
      }
    }
  }
}

// ---------------- Top-16 + scatter-softmax per row ----------------
// gates row = softmax over [4096] vector that is 0 everywhere except top-16 scores.
// m = max(top0, 0); denom = sum exp(topk - m) + (N-16)*exp(-m).
__global__ __launch_bounds__(256)
void topk_softmax(const float* __restrict__ scores, float* __restrict__ gates) {
  __shared__ float srow[NK_];
  __shared__ float rmax[256];
  __shared__ int   ridx[256];
  __shared__ float topv[TOPK_];
  __shared__ int   topi[TOPK_];

  const int row = blockIdx.x;
  const int tid = threadIdx.x;
  const float* s = scores + (size_t)row * NK_;

  for (int i = tid; i < NK_; i += 256) srow[i] = s[i];
  __syncthreads();

  for (int t = 0; t < TOPK_; ++t) {
    float m = -INFINITY;
    int   mi = 0;
    for (int i = tid; i < NK_; i += 256) {
      float v = srow[i];
      if (v > m) { m = v; mi = i; }
    }
    rmax[tid] = m;
    ridx[tid] = mi;
    __syncthreads();
#pragma unroll
    for (int off = 128; off > 0; off >>= 1) {
      if (tid < off) {
        if (rmax[tid + off] > rmax[tid]) {
          rmax[tid] = rmax[tid + off];
          ridx[tid] = ridx[tid + off];
        }
      }
      __syncthreads();
    }
    if (tid == 0) {
      topv[t] = rmax[0];
      topi[t] = ridx[0];
      srow[ridx[0]] = -INFINITY;   // mask out for next round
    }
    __syncthreads();
  }

  // Closed-form softmax of the scattered row (4080 implicit zeros).
  const float m = fmaxf(topv[0], 0.0f);
  float denom = (float)(NK_ - TOPK_) * __expf(-m);
#pragma unroll
  for (int t = 0; t < TOPK_; ++t) denom += __expf(topv[t] - m);
  const float inv = 1.0f / denom;
  const float base = __expf(-m) * inv;

  float* g = gates + (size_t)row * NK_;
  for (int i = tid; i < NK_; i += 256) {
    float v = base;
#pragma unroll
    for (int t = 0; t < TOPK_; ++t)
      if (i == topi[t]) v = __expf(topv[t] - m) * inv;
    g[i] = v;
  }
}

// ---------------- launch ----------------
extern "C" void kernel_launch(void* const* d_in, const int* in_sizes, int n_in,
                              void* d_out, int out_size, void* d_ws, size_t ws_size,
                              hipStream_t stream) {
  const float* x    = (const float*)d_in[0];   // [BS, X_DIM]
  const float* keys = (const float*)d_in[1];   // [N_KEYS, KEY_DIM]
  const float* W    = (const float*)d_in[2];   // [KEY_DIM, X_DIM]
  const float* bias = (const float*)d_in[3];   // [KEY_DIM]
  (void)in_sizes; (void)n_in; (void)out_size; (void)ws_size;

  float* gates  = (float*)d_out;                          // [BS, N_KEYS]
  float* scores = (float*)d_out + (size_t)BS_ * NK_;      // [BS, N_KEYS]
  float* query  = (float*)d_ws;                           // [BS, KEY_DIM] 32 MB scratch

  dim3 blk(256);
  // query = x @ W^T + b : M=16384, N=512, K=4096
  gemm_tn_wmma<<<dim3(BS_ / BM, KD_ / BN), blk, 0, stream>>>(x, W, bias, query,
                                                             BS_, KD_, XD_);
  // scores = query @ keys^T : M=16384, N=4096, K=512
  gemm_tn_wmma<<<dim3(BS_ / BM, NK_ / BN), blk, 0, stream>>>(query, keys, nullptr, scores,
                                                             BS_, NK_, KD_);
  // gates = scatter-softmax(top16(scores))
  topk_softmax<<<dim3(BS_), blk, 0, stream>>>(scores, gates);
}